// DWT_pan_69492570849965
// MI455X (gfx1250) — compile-verified
//
#include <hip/hip_runtime.h>
#include <stdint.h>

// Level-1 2D Haar DWT, 16 x 2048 x 2048 f32 -> (low 16x1x1024x1024, high 16x3x1024x1024).
// Pure HBM-bandwidth-bound (AI ~0.2 FLOP/B -> ~23 us floor at 23.3 TB/s).
// CDNA5 paths: async global->LDS b128 loads (ASYNCcnt) with NT hint + single
// shared address VGPR pair using IOFFSET immediates; non-temporal b128 stores.

typedef float v4f __attribute__((ext_vector_type(4)));

#define W_IN  2048
#define H_IN  2048
#define W_OUT 1024
#define H_OUT 1024

__global__ __launch_bounds__(256) void DWT_haar_gfx1250_kernel(
    const float* __restrict__ in, float* __restrict__ out, int nBatch) {
    // 4 staging slots of 16B per lane: 256 threads * 64B = 16 KB LDS
    __shared__ v4f tile[4 * 256];

    const int tx   = threadIdx.x;
    const int tid  = blockIdx.x * 256 + tx;
    const int colg = tid & 255;          // group of 4 output columns
    const int row  = (tid >> 8) & 1023;  // output row
    const int n    = tid >> 18;          // batch index

    // Input tile start: batch n, row 2*row, col 8*colg
    const size_t inBase = ((size_t)n << 22) + ((size_t)row << 12) + ((size_t)colg << 3);
    const unsigned long long g00 = (unsigned long long)(uintptr_t)(in + inBase);

    // Lane-private LDS byte offsets (low 32 bits of a generic shared pointer
    // are the wave-relative LDS offset on CDNA5).
    const unsigned l0 = (unsigned)(uintptr_t)&tile[tx];
    const unsigned l1 = (unsigned)(uintptr_t)&tile[256 + tx];
    const unsigned l2 = (unsigned)(uintptr_t)&tile[512 + tx];
    const unsigned l3 = (unsigned)(uintptr_t)&tile[768 + tx];

    // CDNA5 async global -> LDS copies (ASYNCcnt). One shared address VGPR
    // pair; row/column deltas via the signed 24-bit IOFFSET immediate:
    //   +16   = row0 cols 4..7
    //   +8192 = row1 cols 0..3 (2048 floats)
    //   +8208 = row1 cols 4..7
    // Single-touch 256 MiB input stream -> non-temporal load hint.
    asm volatile("global_load_async_to_lds_b128 %0, %1, off th:TH_LOAD_NT"
                 :: "v"(l0), "v"(g00) : "memory");
    asm volatile("global_load_async_to_lds_b128 %0, %1, off offset:16 th:TH_LOAD_NT"
                 :: "v"(l1), "v"(g00) : "memory");
    asm volatile("global_load_async_to_lds_b128 %0, %1, off offset:8192 th:TH_LOAD_NT"
                 :: "v"(l2), "v"(g00) : "memory");
    asm volatile("global_load_async_to_lds_b128 %0, %1, off offset:8208 th:TH_LOAD_NT"
                 :: "v"(l3), "v"(g00) : "memory");
    asm volatile("s_wait_asynccnt 0" ::: "memory");

    // Lane reads back exactly the slots it wrote -> no barrier needed.
    const v4f q0 = tile[tx];         // a0 b0 a1 b1   (even input row)
    const v4f q1 = tile[256 + tx];   // a2 b2 a3 b3
    const v4f q2 = tile[512 + tx];   // c0 d0 c1 d1   (odd input row)
    const v4f q3 = tile[768 + tx];   // c2 d2 c3 d3

    const v4f a = { q0.x, q0.z, q1.x, q1.z };   // even row, even col
    const v4f b = { q0.y, q0.w, q1.y, q1.w };   // even row, odd  col
    const v4f c = { q2.x, q2.z, q3.x, q3.z };   // odd  row, even col
    const v4f d = { q2.y, q2.w, q3.y, q3.w };   // odd  row, odd  col

    const v4f sab = a + b, dab = a - b;
    const v4f scd = c + d, dcd = c - d;
    const v4f LL = (sab + scd) * 0.5f;
    const v4f LH = (sab - scd) * 0.5f;
    const v4f HL = (dab + dcd) * 0.5f;
    const v4f HH = (dab - dcd) * 0.5f;

    // Output addressing: low plane then high (LH, HL, HH) channels per batch.
    const size_t opix     = ((size_t)row << 10) + ((size_t)colg << 2);
    float*       lowp     = out + (((size_t)n) << 20) + opix;
    float*       highbase = out + (((size_t)nBatch) << 20);

    // Single-touch 256 MiB output stream vs 192 MB L2 -> non-temporal stores.
    __builtin_nontemporal_store(LL, (v4f*)lowp);
    __builtin_nontemporal_store(LH, (v4f*)(highbase + (((size_t)(3 * n + 0)) << 20) + opix));
    __builtin_nontemporal_store(HL, (v4f*)(highbase + (((size_t)(3 * n + 1)) << 20) + opix));
    __builtin_nontemporal_store(HH, (v4f*)(highbase + (((size_t)(3 * n + 2)) << 20) + opix));
}

extern "C" void kernel_launch(void* const* d_in, const int* in_sizes, int n_in,
                              void* d_out, int out_size, void* d_ws, size_t ws_size,
                              hipStream_t stream) {
    (void)n_in; (void)out_size; (void)d_ws; (void)ws_size;
    const float* in  = (const float*)d_in[0];
    float*       out = (float*)d_out;

    const int nBatch       = in_sizes[0] / (H_IN * W_IN);       // 16
    const int totalThreads = nBatch * H_OUT * (W_OUT / 4);      // 4 output cols / thread
    const int blocks       = totalThreads / 256;                // 16384

    DWT_haar_gfx1250_kernel<<<blocks, 256, 0, stream>>>(in, out, nBatch);
}